// FarthestPointSampler_1717986918813
// MI455X (gfx1250) — compile-verified
//
#include <hip/hip_runtime.h>
#include <float.h>
#include <stdint.h>

// Problem constants (from reference setup_inputs): B=32, C=3, N=131072, S=2048.
#define B_      32
#define N_      131072
#define G_      8              // groups (workgroups) per batch
#define NP_     (N_ / G_)      // 16384 points per group
#define BLOCK_  1024           // 32 waves of 32 (wave32)
#define NWAVES  (BLOCK_ / 32)
#define PPT_    (NP_ / BLOCK_) // 16 points per thread, min-dist held in VGPRs

// 32-byte partial record published by each group each iteration.
struct Part {
    float v;        // group-local max of min_dist
    int   i;        // global index of that point
    float x, y, z;  // its coordinates (so no second exchange round is needed)
    float pad0, pad1, pad2;
};

__global__ void fps_init(int* __restrict__ cnt) {
    if (threadIdx.x < B_) cnt[threadIdx.x] = 0;
}

__launch_bounds__(BLOCK_)
__global__ void fps_kernel(const float* __restrict__ pts_in,   // [B,3,N]
                           const int*   __restrict__ nc_ptr,   // scalar S
                           int*         __restrict__ out,      // [B,S]
                           int*         __restrict__ cnt,      // [B] monotonic arrival counters
                           Part*        __restrict__ parts) {  // [2][B][G] double-buffered
    // 256 KB of dynamic LDS: {x,y,z,pad} per point (coords only; min-dist lives in VGPRs).
    // Needs CDNA5's 320KB-per-WGP LDS.
    extern __shared__ __align__(16) float4 spts[];
    __shared__ float redv[NWAVES];
    __shared__ int   redi[NWAVES];
    __shared__ float bc[3];

    const int g    = blockIdx.x;        // group within batch
    const int b    = blockIdx.y;        // batch
    const int tid  = threadIdx.x;
    const int base = g * NP_;
    const int S    = nc_ptr[0];

    const float* gx = pts_in + (size_t)b * 3 * N_;
    const float* gy = gx + N_;
    const float* gz = gx + 2 * N_;

    // ---- Stage this group's slice into LDS with CDNA5 async global->LDS copies ----
    #pragma unroll
    for (int k = 0; k < PPT_; ++k) {
        const int i = tid + k * BLOCK_;
        uint32_t lx = (uint32_t)(uintptr_t)(&spts[i]);   // low 32 bits of LDS-aperture addr = LDS byte offset
        uint32_t ly = lx + 4;
        uint32_t lz = lx + 8;
        uint64_t ax = (uint64_t)(uintptr_t)(gx + base + i);
        uint64_t ay = (uint64_t)(uintptr_t)(gy + base + i);
        uint64_t az = (uint64_t)(uintptr_t)(gz + base + i);
        asm volatile("global_load_async_to_lds_b32 %0, %1, off" :: "v"(lx), "v"(ax) : "memory");
        asm volatile("global_load_async_to_lds_b32 %0, %1, off" :: "v"(ly), "v"(ay) : "memory");
        asm volatile("global_load_async_to_lds_b32 %0, %1, off" :: "v"(lz), "v"(az) : "memory");
    }
    asm volatile("s_wait_asynccnt 0" ::: "memory");       // wave's own async copies done
    __syncthreads();                                      // -> all waves' copies visible

    // Per-thread register-resident running min squared distance for its 16 fixed points.
    float md[PPT_];
    #pragma unroll
    for (int k = 0; k < PPT_; ++k) md[k] = FLT_MAX;

    // Scan semantics: out[b][0] = 0; step s uses centroid out[b][s-1].
    float cx = gx[0], cy = gy[0], cz = gz[0];
    if (g == 0 && tid == 0) out[(size_t)b * S] = 0;

    const int lane = tid & 31;
    const int wv   = tid >> 5;

    for (int s = 1; s < S; ++s) {
        // ---- update reg-resident min-dist + thread-local argmax (coords from LDS) ----
        float bestv = -1.0f;
        int   besti = 0x7FFFFFFF;
        #pragma unroll
        for (int k = 0; k < PPT_; ++k) {
            const int i = tid + k * BLOCK_;
            float4 p = spts[i];                           // ds_load_b128 (w lane unused)
            float dx = p.x - cx, dy = p.y - cy, dz = p.z - cz;
            float d  = dx * dx + dy * dy + dz * dz;
            float m  = fminf(md[k], d);
            md[k] = m;                                    // stays in VGPR
            if (m > bestv) { bestv = m; besti = base + i; }  // strict >: lowest index on ties
        }

        // ---- wave32 shuffle reduction (tie -> lower index, matching jnp.argmax) ----
        for (int off = 16; off > 0; off >>= 1) {
            float ov = __shfl_down(bestv, off, 32);
            int   oi = __shfl_down(besti, off, 32);
            if (ov > bestv || (ov == bestv && oi < besti)) { bestv = ov; besti = oi; }
        }
        if (lane == 0) { redv[wv] = bestv; redi[wv] = besti; }
        __syncthreads();

        if (wv == 0) {
            bestv = (lane < NWAVES) ? redv[lane] : -1.0f;
            besti = (lane < NWAVES) ? redi[lane] : 0x7FFFFFFF;
            for (int off = 16; off > 0; off >>= 1) {
                float ov = __shfl_down(bestv, off, 32);
                int   oi = __shfl_down(besti, off, 32);
                if (ov > bestv || (ov == bestv && oi < besti)) { bestv = ov; besti = oi; }
            }
            if (lane == 0) {
                // Winner of this group lives in this group's LDS slice: grab its coords.
                float4 w = spts[besti - base];
                Part* mp = &parts[((size_t)(s & 1) * B_ + b) * G_ + g];
                mp->v = bestv; mp->i = besti; mp->x = w.x; mp->y = w.y; mp->z = w.z;
                __threadfence();
                __hip_atomic_fetch_add(&cnt[b], 1, __ATOMIC_RELEASE, __HIP_MEMORY_SCOPE_AGENT);

                // Spin until all G groups of this batch arrived for step s (monotonic target).
                const int target = G_ * s;
                while (__hip_atomic_load(&cnt[b], __ATOMIC_ACQUIRE, __HIP_MEMORY_SCOPE_AGENT) < target) {
                    __builtin_amdgcn_s_sleep(1);
                }

                // All-to-all: every group reduces the same 8 records identically.
                const Part* pb = &parts[((size_t)(s & 1) * B_ + b) * G_];
                float wvv = -1.0f, wx = 0.f, wy = 0.f, wz = 0.f;
                int   wii = 0x7FFFFFFF;
                #pragma unroll
                for (int q = 0; q < G_; ++q) {
                    float v = pb[q].v; int ii = pb[q].i;
                    if (v > wvv || (v == wvv && ii < wii)) {
                        wvv = v; wii = ii; wx = pb[q].x; wy = pb[q].y; wz = pb[q].z;
                    }
                }
                bc[0] = wx; bc[1] = wy; bc[2] = wz;
                if (g == 0) out[(size_t)b * S + s] = wii;
            }
        }
        __syncthreads();
        cx = bc[0]; cy = bc[1]; cz = bc[2];
    }
}

extern "C" void kernel_launch(void* const* d_in, const int* in_sizes, int n_in,
                              void* d_out, int out_size, void* d_ws, size_t ws_size,
                              hipStream_t stream) {
    (void)in_sizes; (void)n_in; (void)out_size; (void)ws_size;
    const float* pts = (const float*)d_in[0];
    const int*   nc  = (const int*)d_in[1];     // num_centroids scalar (2048)
    int*         out = (int*)d_out;             // int32 indices [B, S]

    int*  cnt   = (int*)d_ws;                    // [B] counters at offset 0
    Part* parts = (Part*)((char*)d_ws + 256);    // [2][B][G] records (16 KB)

    fps_init<<<1, 64, 0, stream>>>(cnt);         // zero counters every launch (ws is poisoned)
    fps_kernel<<<dim3(G_, B_), BLOCK_, NP_ * sizeof(float4), stream>>>(pts, nc, out, cnt, parts);
}